// SubGraphModel_1142461300968
// MI455X (gfx1250) — compile-verified
//
#include <hip/hip_runtime.h>
#include <hip/hip_bf16.h>

// Problem constants (fixed by the reference).
#define NN 1048576
#define EE 8388608
#define GG 32768

typedef float v2f __attribute__((ext_vector_type(2)));
typedef float v8f __attribute__((ext_vector_type(8)));

__device__ __forceinline__ void gAtomAdd(float* p, float v) {
  __hip_atomic_fetch_add(p, v, __ATOMIC_RELAXED, __HIP_MEMORY_SCOPE_AGENT);
}

// ---------------- utility kernels ----------------
__global__ void k_zero(float* p, int cnt) {
  int i = blockIdx.x * blockDim.x + threadIdx.x;
  if (i < cnt) p[i] = 0.f;
}

__global__ void k_copy(float* dst, const float* src, int cnt) {
  int i = blockIdx.x * blockDim.x + threadIdx.x;
  if (i < cnt) dst[i] = src[i];
}

// ---------------- atom encoder: h[n,0..8] = sum_f atom_emb[f, x[n,f], :] ----
__global__ void k_atom(const int* __restrict__ x, const float* __restrict__ ae,
                       float* __restrict__ h, int n) {
  int i = blockIdx.x * blockDim.x + threadIdx.x;
  if (i >= n) return;
  float acc[9];
#pragma unroll
  for (int d = 0; d < 9; d++) acc[d] = 0.f;
#pragma unroll
  for (int f = 0; f < 9; f++) {
    int v = x[(size_t)i * 9 + f];
    const float* row = ae + ((size_t)f * 119 + v) * 9;
#pragma unroll
    for (int d = 0; d < 9; d++) acc[d] += row[d];
  }
  float* hr = h + (size_t)i * 12;
#pragma unroll
  for (int d = 0; d < 9; d++) hr[d] = acc[d];
  hr[9] = 0.f; hr[10] = 0.f; hr[11] = 0.f;
}

// ---------------- per-layer node prep: h_in = h + vn[batch]; zpre = (1+eps)h_in
__global__ void k_prepare(const float* __restrict__ h, const float* __restrict__ vn,
                          const int* __restrict__ batch, const float* __restrict__ eps_ptr,
                          float* __restrict__ h_in, float* __restrict__ zpre, int n) {
  int i = blockIdx.x * blockDim.x + threadIdx.x;
  if (i >= n) return;
  float e1 = 1.0f + *eps_ptr;
  int g = batch[i];
  const float* hv = h + (size_t)i * 12;
  const float* vv = vn + (size_t)g * 12;
  float* hi = h_in + (size_t)i * 12;
  float* zp = zpre + (size_t)i * 12;
#pragma unroll
  for (int d = 0; d < 9; d++) {
    float xv = hv[d] + vv[d];
    hi[d] = xv;
    zp[d] = e1 * xv;
  }
#pragma unroll
  for (int d = 9; d < 12; d++) { hi[d] = 0.f; zp[d] = 0.f; }
}

// ---------------- edge kernel: zpre[dst] += relu(h_in[src] + bond_emb_sum) ---
__global__ void k_edge(const int* __restrict__ ei, const int* __restrict__ ea,
                       const float* __restrict__ be, const float* __restrict__ h_in,
                       float* __restrict__ zpre, int ne) {
  int e = blockIdx.x * blockDim.x + threadIdx.x;
  if (e >= ne) return;
  int s = ei[e];
  int d = ei[ne + e];
  int a0 = ea[(size_t)e * 3 + 0];
  int a1 = ea[(size_t)e * 3 + 1];
  int a2 = ea[(size_t)e * 3 + 2];
  const float* b0 = be + (0 * 6 + a0) * 9;
  const float* b1 = be + (1 * 6 + a1) * 9;
  const float* b2 = be + (2 * 6 + a2) * 9;
  const float4* hr = (const float4*)(h_in + (size_t)s * 12);
  float4 h0 = hr[0], h1 = hr[1], h2 = hr[2];
  float hv[12] = {h0.x, h0.y, h0.z, h0.w, h1.x, h1.y, h1.z, h1.w,
                  h2.x, h2.y, h2.z, h2.w};
  float* zr = zpre + (size_t)d * 12;
#pragma unroll
  for (int j = 0; j < 9; j++) {
    float m = hv[j] + b0[j] + b1[j] + b2[j];
    m = m > 0.f ? m : 0.f;
    gAtomAdd(&zr[j], m);
  }
}

// ---------------- weight prep: pad W1(9x18)->[12x32], W2(18x9)->[20x16] ------
__global__ void k_pack(const float* __restrict__ W1s, const float* __restrict__ b1s,
                       const float* __restrict__ W2s, const float* __restrict__ b2s,
                       float* Wp1, float* bp1, float* Wp2, float* bp2,
                       float* S1, float* Q1, float* S2, float* Q2) {
  int t = threadIdx.x;  // 512 threads
  if (t < 384) Wp1[t] = 0.f;
  if (t < 320) Wp2[t] = 0.f;
  if (t < 32) { bp1[t] = 0.f; S1[t] = 0.f; Q1[t] = 0.f; }
  if (t < 16) { bp2[t] = 0.f; S2[t] = 0.f; Q2[t] = 0.f; }
  __syncthreads();
  if (t < 162) { int k = t / 18, nn = t % 18; Wp1[k * 32 + nn] = W1s[t]; }
  if (t >= 256 && t < 256 + 162) {
    int u = t - 256; int k = u / 9, nn = u % 9; Wp2[k * 16 + nn] = W2s[u];
  }
  if (t < 18) bp1[t] = b1s[t];
  if (t >= 32 && t < 41) bp2[t - 32] = b2s[t - 32];
}

// ---------------- GEMM1 via WMMA f32 16x16x4: T[rows,20] = A[rows,12] @ Wp1[12,32] + bias
// also accumulates per-column sum / sumsq for BatchNorm.
__global__ void k_gemm1(const float* __restrict__ A, const float* __restrict__ Wp,
                        const float* __restrict__ bp, float* __restrict__ T,
                        float* __restrict__ S, float* __restrict__ Q, int rows) {
  __shared__ float sS[32], sQ[32];
  int tid = threadIdx.x;
  if (tid < 32) { sS[tid] = 0.f; sQ[tid] = 0.f; }
  __syncthreads();
  int wave = tid >> 5, lane = tid & 31;
  int tile = blockIdx.x * 8 + wave;
  int row0 = tile * 16;
  int col = lane & 15;
  int kgrp = (lane >> 4) << 1;  // 0 or 2
  int arow = row0 + col;        // A-fragment: M = lane%16, K = 4c + kgrp + v
  v2f a0 = *(const v2f*)(A + (size_t)arow * 12 + 0 + kgrp);
  v2f a1 = *(const v2f*)(A + (size_t)arow * 12 + 4 + kgrp);
  v2f a2 = *(const v2f*)(A + (size_t)arow * 12 + 8 + kgrp);
  v8f acc0, acc1;
  float bias0 = bp[col], bias1 = bp[16 + col];
#pragma unroll
  for (int r = 0; r < 8; r++) { acc0[r] = bias0; acc1[r] = bias1; }
#pragma unroll
  for (int c = 0; c < 3; c++) {
    v2f a = (c == 0) ? a0 : ((c == 1) ? a1 : a2);
    int kb = c * 4 + kgrp;
    v2f b0, b1;
    b0.x = Wp[kb * 32 + col];       b0.y = Wp[(kb + 1) * 32 + col];
    b1.x = Wp[kb * 32 + 16 + col];  b1.y = Wp[(kb + 1) * 32 + 16 + col];
    acc0 = __builtin_amdgcn_wmma_f32_16x16x4_f32(false, a, false, b0, (short)0,
                                                 acc0, false, false);
    acc1 = __builtin_amdgcn_wmma_f32_16x16x4_f32(false, a, false, b1, (short)0,
                                                 acc1, false, false);
  }
  int rbase = row0 + ((lane >> 4) << 3);  // rows r + 8*(lane/16)
  // Straight-line coalesced stores for columns 0..15 + stats for both tiles.
  float s0 = 0.f, q0 = 0.f, s1 = 0.f, q1 = 0.f;
#pragma unroll
  for (int r = 0; r < 8; r++) {
    float v0 = acc0[r], v1 = acc1[r];
    T[(size_t)(rbase + r) * 20 + col] = v0;
    s0 += v0; q0 += v0 * v0; s1 += v1; q1 += v1 * v1;
  }
  // Single diverged region for columns 16..19 (18,19 store exact zeros).
  if (col < 4) {
#pragma unroll
    for (int r = 0; r < 8; r++) {
      T[(size_t)(rbase + r) * 20 + 16 + col] = acc1[r];
    }
  }
  atomicAdd(&sS[col], s0);      atomicAdd(&sQ[col], q0);
  atomicAdd(&sS[16 + col], s1); atomicAdd(&sQ[16 + col], q1);
  __syncthreads();
  if (tid < 32) { gAtomAdd(&S[tid], sS[tid]); gAtomAdd(&Q[tid], sQ[tid]); }
}

// ---------------- BN stat finalize: scale = g*rsqrt(var+eps), shift = b - m*scale
__global__ void k_finalize(const float* __restrict__ S, const float* __restrict__ Q,
                           const float* __restrict__ g, const float* __restrict__ b,
                           float* sc, float* sh, float inv_count, int nvalid, int npad) {
  int c = threadIdx.x;
  if (c < npad) {
    if (c < nvalid) {
      float m = S[c] * inv_count;
      float var = Q[c] * inv_count - m * m;
      float r = rsqrtf(var + 1e-5f);
      float scale = g[c] * r;
      sc[c] = scale;
      sh[c] = b[c] - m * scale;
    } else {
      sc[c] = 0.f; sh[c] = 0.f;
    }
  }
}

// ---------------- GEMM2 via WMMA: T2[rows,16] = relu(T1*scale+shift)[rows,20] @ Wp2[20,16] + bias
__global__ void k_gemm2(const float* __restrict__ T1, const float* __restrict__ sc,
                        const float* __restrict__ sh, const float* __restrict__ Wp,
                        const float* __restrict__ bp, float* __restrict__ T2,
                        float* __restrict__ S, float* __restrict__ Q, int rows) {
  __shared__ float sS[16], sQ[16];
  int tid = threadIdx.x;
  if (tid < 16) { sS[tid] = 0.f; sQ[tid] = 0.f; }
  __syncthreads();
  int wave = tid >> 5, lane = tid & 31;
  int tile = blockIdx.x * 8 + wave;
  int row0 = tile * 16;
  int col = lane & 15;
  int kgrp = (lane >> 4) << 1;
  int arow = row0 + col;
  v8f acc;
  float bias = bp[col];
#pragma unroll
  for (int r = 0; r < 8; r++) acc[r] = bias;
#pragma unroll
  for (int c = 0; c < 5; c++) {
    int k0 = c * 4 + kgrp;
    v2f tv = *(const v2f*)(T1 + (size_t)arow * 20 + k0);
    v2f a;
    a.x = fmaxf(tv.x * sc[k0] + sh[k0], 0.f);
    a.y = fmaxf(tv.y * sc[k0 + 1] + sh[k0 + 1], 0.f);
    v2f b;
    b.x = Wp[k0 * 16 + col];
    b.y = Wp[(k0 + 1) * 16 + col];
    acc = __builtin_amdgcn_wmma_f32_16x16x4_f32(false, a, false, b, (short)0,
                                                acc, false, false);
  }
  int rbase = row0 + ((lane >> 4) << 3);
  float s = 0.f, q = 0.f;
#pragma unroll
  for (int r = 0; r < 8; r++) {
    float v = acc[r];
    T2[(size_t)(rbase + r) * 16 + col] = v;  // cols 9..15 are exact zeros
    s += v; q += v * v;
  }
  atomicAdd(&sS[col], s);
  atomicAdd(&sQ[col], q);
  __syncthreads();
  if (tid < 16) { gAtomAdd(&S[tid], sS[tid]); gAtomAdd(&Q[tid], sQ[tid]); }
}

// ---------------- node apply: h = (relu?)(BN(t2)) + h_in ---------------------
__global__ void k_apply_node(const float* __restrict__ t2, const float* __restrict__ sc,
                             const float* __restrict__ sh, const float* __restrict__ h_in,
                             float* __restrict__ h, int n, int do_relu) {
  int i = blockIdx.x * blockDim.x + threadIdx.x;
  if (i >= n) return;
  const float* t = t2 + (size_t)i * 16;
  const float* hi = h_in + (size_t)i * 12;
  float* hr = h + (size_t)i * 12;
#pragma unroll
  for (int d = 0; d < 9; d++) {
    float z = t[d] * sc[d] + sh[d];
    if (do_relu) z = fmaxf(z, 0.f);
    hr[d] = z + hi[d];
  }
  hr[9] = 0.f; hr[10] = 0.f; hr[11] = 0.f;
}

// ---------------- virtual node: vn_tmp[batch] += h_in ------------------------
__global__ void k_vn_accum(const float* __restrict__ h_in, const int* __restrict__ batch,
                           float* __restrict__ vn_tmp, int n) {
  int i = blockIdx.x * blockDim.x + threadIdx.x;
  if (i >= n) return;
  int g = batch[i];
  const float* hi = h_in + (size_t)i * 12;
  float* vt = vn_tmp + (size_t)g * 12;
#pragma unroll
  for (int d = 0; d < 9; d++) gAtomAdd(&vt[d], hi[d]);
}

__global__ void k_vn_apply(const float* __restrict__ vt2, const float* __restrict__ sc,
                           const float* __restrict__ sh, float* __restrict__ vn, int ng) {
  int i = blockIdx.x * blockDim.x + threadIdx.x;
  if (i >= ng) return;
  const float* t = vt2 + (size_t)i * 16;
  float* v = vn + (size_t)i * 12;
#pragma unroll
  for (int d = 0; d < 9; d++) v[d] += fmaxf(t[d] * sc[d] + sh[d], 0.f);
}

// ---------------- final reduce: out[0,:9] = sum_n h[n,:9] --------------------
__global__ void k_final(const float* __restrict__ h, float* __restrict__ out, int n) {
  __shared__ float s[9];
  if (threadIdx.x < 9) s[threadIdx.x] = 0.f;
  __syncthreads();
  float acc[9];
#pragma unroll
  for (int d = 0; d < 9; d++) acc[d] = 0.f;
  for (int i = blockIdx.x * blockDim.x + threadIdx.x; i < n;
       i += gridDim.x * blockDim.x) {
    const float* hr = h + (size_t)i * 12;
#pragma unroll
    for (int d = 0; d < 9; d++) acc[d] += hr[d];
  }
#pragma unroll
  for (int d = 0; d < 9; d++) atomicAdd(&s[d], acc[d]);
  __syncthreads();
  if (threadIdx.x < 9) gAtomAdd(&out[threadIdx.x], s[threadIdx.x]);
}

extern "C" void kernel_launch(void* const* d_in, const int* in_sizes, int n_in,
                              void* d_out, int out_size, void* d_ws, size_t ws_size,
                              hipStream_t stream) {
  const float* atom_emb = (const float*)d_in[0];   // (9,119,9)
  const float* bond_emb = (const float*)d_in[1];   // (3,3,6,9)
  const float* eps_gin  = (const float*)d_in[2];   // (3,)
  const float* W1    = (const float*)d_in[3];      // (3,9,18)
  const float* b1    = (const float*)d_in[4];
  const float* bn1_g = (const float*)d_in[5];
  const float* bn1_b = (const float*)d_in[6];
  const float* W2    = (const float*)d_in[7];      // (3,18,9)
  const float* b2    = (const float*)d_in[8];
  const float* bno_g = (const float*)d_in[9];
  const float* bno_b = (const float*)d_in[10];
  const float* vnW1    = (const float*)d_in[11];   // (2,9,18)
  const float* vnb1    = (const float*)d_in[12];
  const float* vnbn1_g = (const float*)d_in[13];
  const float* vnbn1_b = (const float*)d_in[14];
  const float* vnW2    = (const float*)d_in[15];   // (2,18,9)
  const float* vnb2    = (const float*)d_in[16];
  const float* vnbn2_g = (const float*)d_in[17];
  const float* vnbn2_b = (const float*)d_in[18];
  const int* x     = (const int*)d_in[19];         // (N,9)
  const int* ei    = (const int*)d_in[20];         // (2,E)
  const int* ea    = (const int*)d_in[21];         // (E,3)
  const int* batch = (const int*)d_in[22];         // (N,)
  float* out = (float*)d_out;

  float* ws = (float*)d_ws;
  // Workspace layout (floats). zpre aliases t2 (non-overlapping lifetimes).
  size_t oH   = 0;                          // h     [N,12]
  size_t oHin = oH   + (size_t)NN * 12;     // h_in  [N,12]
  size_t oT1  = oHin + (size_t)NN * 12;     // t1    [N,20]
  size_t oT2  = oT1  + (size_t)NN * 20;     // t2 [N,16] / zpre [N,12]
  size_t oVN  = oT2  + (size_t)NN * 16;     // vn    [G,12]
  size_t oVNT = oVN  + (size_t)GG * 12;     // vn_tmp[G,12]
  size_t oVT1 = oVNT + (size_t)GG * 12;     // vt1   [G,20]
  size_t oVT2 = oVT1 + (size_t)GG * 20;     // vt2   [G,16]
  size_t oWp1 = oVT2 + (size_t)GG * 16;     // 12*32
  size_t oBp1 = oWp1 + 384;                 // 32
  size_t oWp2 = oBp1 + 32;                  // 20*16
  size_t oBp2 = oWp2 + 320;                 // 16
  size_t oS1  = oBp2 + 16;
  size_t oQ1  = oS1 + 32;
  size_t oSc1 = oQ1 + 32;
  size_t oSh1 = oSc1 + 32;
  size_t oS2  = oSh1 + 32;
  size_t oQ2  = oS2 + 16;
  size_t oSc2 = oQ2 + 16;
  size_t oSh2 = oSc2 + 16;

  float *h = ws + oH, *h_in = ws + oHin, *t1 = ws + oT1, *t2 = ws + oT2;
  float *zpre = ws + oT2;  // alias
  float *vn = ws + oVN, *vn_tmp = ws + oVNT, *vt1 = ws + oVT1, *vt2 = ws + oVT2;
  float *Wp1 = ws + oWp1, *bp1 = ws + oBp1, *Wp2 = ws + oWp2, *bp2 = ws + oBp2;
  float *S1 = ws + oS1, *Q1 = ws + oQ1, *Sc1 = ws + oSc1, *Sh1 = ws + oSh1;
  float *S2 = ws + oS2, *Q2 = ws + oQ2, *Sc2 = ws + oSc2, *Sh2 = ws + oSh2;

  const float invN = 1.0f / (float)NN;
  const float invG = 1.0f / (float)GG;

  k_zero<<<(GG * 12) / 256, 256, 0, stream>>>(vn, GG * 12);
  k_atom<<<NN / 256, 256, 0, stream>>>(x, atom_emb, h, NN);

  for (int l = 0; l < 3; l++) {
    k_prepare<<<NN / 256, 256, 0, stream>>>(h, vn, batch, eps_gin + l, h_in, zpre, NN);
    k_edge<<<EE / 256, 256, 0, stream>>>(ei, ea, bond_emb + (size_t)l * 162, h_in, zpre, EE);
    k_pack<<<1, 512, 0, stream>>>(W1 + l * 162, b1 + l * 18, W2 + l * 162, b2 + l * 9,
                                  Wp1, bp1, Wp2, bp2, S1, Q1, S2, Q2);
    k_gemm1<<<NN / 128, 256, 0, stream>>>(zpre, Wp1, bp1, t1, S1, Q1, NN);
    k_finalize<<<1, 32, 0, stream>>>(S1, Q1, bn1_g + l * 18, bn1_b + l * 18,
                                     Sc1, Sh1, invN, 18, 32);
    k_gemm2<<<NN / 128, 256, 0, stream>>>(t1, Sc1, Sh1, Wp2, bp2, t2, S2, Q2, NN);
    k_finalize<<<1, 32, 0, stream>>>(S2, Q2, bno_g + l * 9, bno_b + l * 9,
                                     Sc2, Sh2, invN, 9, 16);
    k_apply_node<<<NN / 256, 256, 0, stream>>>(t2, Sc2, Sh2, h_in, h, NN, (l < 2) ? 1 : 0);

    if (l < 2) {
      k_copy<<<(GG * 12) / 256, 256, 0, stream>>>(vn_tmp, vn, GG * 12);
      k_vn_accum<<<NN / 256, 256, 0, stream>>>(h_in, batch, vn_tmp, NN);
      k_pack<<<1, 512, 0, stream>>>(vnW1 + l * 162, vnb1 + l * 18, vnW2 + l * 162,
                                    vnb2 + l * 9, Wp1, bp1, Wp2, bp2, S1, Q1, S2, Q2);
      k_gemm1<<<GG / 128, 256, 0, stream>>>(vn_tmp, Wp1, bp1, vt1, S1, Q1, GG);
      k_finalize<<<1, 32, 0, stream>>>(S1, Q1, vnbn1_g + l * 18, vnbn1_b + l * 18,
                                       Sc1, Sh1, invG, 18, 32);
      k_gemm2<<<GG / 128, 256, 0, stream>>>(vt1, Sc1, Sh1, Wp2, bp2, vt2, S2, Q2, GG);
      k_finalize<<<1, 32, 0, stream>>>(S2, Q2, vnbn2_g + l * 9, vnbn2_b + l * 9,
                                       Sc2, Sh2, invG, 9, 16);
      k_vn_apply<<<GG / 256, 256, 0, stream>>>(vt2, Sc2, Sh2, vn, GG);
    }
  }

  k_zero<<<1, 32, 0, stream>>>(out, 9);
  k_final<<<1024, 256, 0, stream>>>(h, out, NN);
}